// Transducer_86071144612049
// MI455X (gfx1250) — compile-verified
//
#include <hip/hip_runtime.h>
#include <math.h>

typedef float v2f __attribute__((ext_vector_type(2)));
typedef float v8f __attribute__((ext_vector_type(8)));

#define B_   128
#define T_   2000
#define C_   62
#define L_   256
#define NEGV (-1e30f)

// ---------------------------------------------------------------------------
// Kernel 1: lse[b*T+t] = log(sum_c exp(x[b,t,c]))
// One wave32 computes 16 rows' sums via V_WMMA_F32_16X16X4_F32 (exact f32):
//   A = 16x4 tile of exp(x), B = 4x16 ones, accumulate over 16 K-chunks (C
//   padded 62->64 with zeros). Every column of D then holds the row sum.
// ---------------------------------------------------------------------------
__global__ __launch_bounds__(256) void lse_kernel(const float* __restrict__ x,
                                                  float* __restrict__ lse) {
  const int lane    = threadIdx.x & 31;
  const int wid     = threadIdx.x >> 5;
  const int rowBase = blockIdx.x * 128 + wid * 16;   // 16 rows per wave
  const int mrow    = lane & 15;
  const int half    = lane >> 4;                     // 0: K=0,1  1: K=2,3
  const float* __restrict__ xr = x + (long)(rowBase + mrow) * C_;

  v8f acc  = {0.f, 0.f, 0.f, 0.f, 0.f, 0.f, 0.f, 0.f};
  v2f ones = {1.f, 1.f};

#pragma unroll
  for (int k = 0; k < 64; k += 4) {
    const int c0 = k + half * 2;
    const int c1 = c0 + 1;
    const float x0 = xr[c0 < C_ ? c0 : (C_ - 1)];    // clamp: avoid OOB read
    const float x1 = xr[c1 < C_ ? c1 : (C_ - 1)];
    v2f a;
    a.x = (c0 < C_) ? __expf(x0) : 0.f;              // pad c=62,63 with 0
    a.y = (c1 < C_) ? __expf(x1) : 0.f;
    // 8 args: (neg_a, A, neg_b, B, c_mod, C, reuse_a, reuse_b)
    acc = __builtin_amdgcn_wmma_f32_16x16x4_f32(false, a, false, ones,
                                                (short)0, acc, false, false);
  }

  // D layout: VGPR v, lanes 0-15 -> M=v ; lanes 16-31 -> M=v+8.
  if (lane == 0) {
#pragma unroll
    for (int i = 0; i < 8; ++i) lse[rowBase + i] = __logf(acc[i]);
  } else if (lane == 16) {
#pragma unroll
    for (int i = 0; i < 8; ++i) lse[rowBase + 8 + i] = __logf(acc[i]);
  }
}

// ---------------------------------------------------------------------------
// Kernel 2: the serial-in-T DP. One block per batch element, thread u holds
// alpha[b,u]. Neighbor alpha[u-1] via double-buffered LDS (one barrier per
// step); next step's emission gather is issued before the barrier to hide
// L2 latency behind the sync.
// ---------------------------------------------------------------------------
__global__ __launch_bounds__(L_) void dp_kernel(const float* __restrict__ x,
                                                const float* __restrict__ lse,
                                                const int* __restrict__ tgts,
                                                float* __restrict__ partial) {
  const int b = blockIdx.x;
  const int u = threadIdx.x;
  const int tgt = tgts[b * L_ + u];
  const float* __restrict__ xb   = x   + (size_t)b * T_ * C_;
  const float* __restrict__ lseb = lse + (size_t)b * T_;

  __shared__ float buf[2][L_];

  float alpha = (u == 0) ? (xb[tgt] - lseb[0]) : NEGV;
  float emv = xb[C_ + tgt] - lseb[1];                 // emission for t=1

  for (int t = 1; t < T_; ++t) {
    buf[t & 1][u] = alpha;
    float em_nxt = 0.f;
    if (t + 1 < T_)                                   // prefetch next emission
      em_nxt = xb[(size_t)(t + 1) * C_ + tgt] - lseb[t + 1];
    __syncthreads();
    const float prev = (u == 0) ? NEGV : buf[t & 1][u - 1];
    const float mx = fmaxf(alpha, prev);
    const float dl = fminf(alpha, prev) - mx;         // <= 0
    alpha = mx + log1pf(__expf(dl)) + emv;            // logaddexp + em
    emv = em_nxt;
  }

  if (u == L_ - 1) partial[b] = -alpha;
}

// ---------------------------------------------------------------------------
// Kernel 3: deterministic mean over the 128 per-batch losses (fixed tree,
// no float atomics -> identical result on every replay).
// ---------------------------------------------------------------------------
__global__ void reduce_kernel(const float* __restrict__ partial,
                              float* __restrict__ out) {
  const int lane = threadIdx.x;                       // 32 threads, one wave
  float s = partial[lane] + partial[lane + 32] +
            partial[lane + 64] + partial[lane + 96];
#pragma unroll
  for (int off = 16; off > 0; off >>= 1)
    s += __shfl_down(s, off, 32);
  if (lane == 0) out[0] = s * (1.0f / (float)B_);
}

extern "C" void kernel_launch(void* const* d_in, const int* in_sizes, int n_in,
                              void* d_out, int out_size, void* d_ws, size_t ws_size,
                              hipStream_t stream) {
  const float* x  = (const float*)d_in[0];   // [B, T, C] float32
  const int*   tg = (const int*)d_in[1];     // [B, L]    int32
  float* lse     = (float*)d_ws;             // B*T floats
  float* partial = lse + (size_t)B_ * T_;    // B floats
  float* out     = (float*)d_out;            // 1 float

  lse_kernel<<<(B_ * T_) / 128, 256, 0, stream>>>(x, lse);
  dp_kernel<<<B_, L_, 0, stream>>>(x, lse, tg, partial);
  reduce_kernel<<<1, 32, 0, stream>>>(partial, out);
}